// LUTFakeQuant_85590108274702
// MI455X (gfx1250) — compile-verified
//
#include <hip/hip_runtime.h>

// LUT fake-quant, MI455X (gfx1250).
//
// Roofline: 12.85M fp32 elems, 103MB traffic @ 23.3 TB/s => ~4.4us floor. No matmul
// structure -> WMMA inapplicable; goal is HBM saturation with minimal VALU/elem.
//   - global_load_async_to_lds_b128 double-buffered staging (ASYNCcnt pipeline, CDNA5)
//   - sorted-centers + midpoint cndmask ladder: 15 cmp + 15 sel, no DS gather
//   - v_med3_f32 clamp, register-resident per-channel constants
//   - nontemporal b128 stores (write-once output)

#define TPB 256
#define KC  16

typedef float vf4 __attribute__((ext_vector_type(4)));   // native vector: b128 + NT builtin OK

__global__ __launch_bounds__(TPB) void lutfq_kernel(
    const float* __restrict__ x,
    const float* __restrict__ scales,
    const float* __restrict__ centers,
    float* __restrict__ out,
    int n4, int nTiles)
{
    __shared__ vf4   stage[2][TPB];       // 8 KB async staging, double buffered
    __shared__ float s_sorted[KC];        // sorted rounded centers

    const int tid = threadIdx.x;

    // LDS byte offsets for the async destination (low 32 bits of generic LDS ptr)
    const unsigned lds0 = (unsigned)(unsigned long long)(void*)&stage[0][tid];
    const unsigned lds1 = (unsigned)(unsigned long long)(void*)&stage[1][tid];

    const vf4* __restrict__ x4 = (const vf4*)x;
    vf4* __restrict__ o4 = (vf4*)out;

    int tile = blockIdx.x;
    if (tile >= nTiles) return;

    // ---- prologue: kick off first async tile load immediately (overlaps setup) ----
    {
        int f4 = tile * TPB + tid;
        if (f4 > n4 - 1) f4 = n4 - 1;              // clamp (shape divides exactly; safety)
        unsigned long long ga = (unsigned long long)(x4 + f4);
        asm volatile("global_load_async_to_lds_b128 %0, %1, off"
                     :: "v"(lds0), "v"(ga) : "memory");
    }

    // ---- setup: rank-sort the 16 rounded centers (uniform data, O(K^2) rank) ----
    if (tid < KC) {
        float ci = rintf(centers[tid]);            // jnp.round = half-even
        int rank = 0;
        #pragma unroll
        for (int j = 0; j < KC; ++j) {
            float cj = rintf(centers[j]);
            rank += (cj < ci) || (cj == ci && j < tid);
        }
        s_sorted[rank] = ci;
    }
    __syncthreads();

    // sorted centers + midpoints -> registers (one-time; reused every element)
    float sc[KC];
    #pragma unroll
    for (int i = 0; i < KC; ++i) sc[i] = s_sorted[i];
    float mid[KC - 1];
    #pragma unroll
    for (int i = 0; i < KC - 1; ++i) mid[i] = 0.5f * (sc[i] + sc[i + 1]);

    // per-thread channel constants: tile = 256 float4 = 1024 elems = 4 channel cycles
    // channel of elem (f4*4+j) mod 256 == (tid&63)*4 + j for every tile
    const int cb = (tid & 63) * 4;
    float inv[4], osc[4];
    #pragma unroll
    for (int j = 0; j < 4; ++j) {
        float s = scales[cb + j];
        inv[j] = 128.0f / (s + 1e-8f);             // x -> t scale
        osc[j] = s * (1.0f / 128.0f);              // center -> out scale
    }

    // ---- main pipeline: issue tile n+1, wait tile n, compute tile n ----
    int buf = 0;
    while (tile < nTiles) {
        const int next = tile + (int)gridDim.x;    // uniform branch
        if (next < nTiles) {
            int f4 = next * TPB + tid;
            if (f4 > n4 - 1) f4 = n4 - 1;
            unsigned long long ga = (unsigned long long)(x4 + f4);
            const unsigned dst = buf ? lds0 : lds1;
            asm volatile("global_load_async_to_lds_b128 %0, %1, off"
                         :: "v"(dst), "v"(ga) : "memory");
            asm volatile("s_wait_asynccnt 0x1" ::: "memory");   // current tile done
        } else {
            asm volatile("s_wait_asynccnt 0x0" ::: "memory");
        }

        const vf4 v = stage[buf][tid];             // ds_load_b128 (own slot; no barrier)
        float r[4] = { v.x, v.y, v.z, v.w };
        float res[4];
        #pragma unroll
        for (int j = 0; j < 4; ++j) {
            // v_mul + v_med3_f32 clamp
            float t = __builtin_amdgcn_fmed3f(r[j] * inv[j], -128.0f, 127.0f);
            // nearest sorted center via 15x (v_cmp + v_cndmask); no memory op
            float c = sc[0];
            #pragma unroll
            for (int i = 0; i < KC - 1; ++i)
                c = (t > mid[i]) ? sc[i + 1] : c;
            res[j] = c * osc[j];
        }

        const int f4 = tile * TPB + tid;
        if (f4 < n4) {
            vf4 o;
            o.x = res[0]; o.y = res[1]; o.z = res[2]; o.w = res[3];
            __builtin_nontemporal_store(o, &o4[f4]); // b128 NT store, write-once data
        }

        tile = next;
        buf ^= 1;
    }
}

extern "C" void kernel_launch(void* const* d_in, const int* in_sizes, int n_in,
                              void* d_out, int out_size, void* d_ws, size_t ws_size,
                              hipStream_t stream) {
    const float* x       = (const float*)d_in[0];   // [16,56,56,256] fp32
    const float* scales  = (const float*)d_in[1];   // [256] fp32
    const float* centers = (const float*)d_in[2];   // [16]  fp32
    float* out = (float*)d_out;

    const int n  = in_sizes[0];                     // 12,845,056
    const int n4 = n / 4;                           // 3,211,264 float4
    const int nTiles = (n4 + TPB - 1) / TPB;        // 12,544 tiles of 256 float4

    int blocks = nTiles < 1568 ? nTiles : 1568;     // ~8 tiles/block: amortize setup,
    if (blocks < 1) blocks = 1;                     // still ~12.5K waves of parallelism

    lutfq_kernel<<<blocks, TPB, 0, stream>>>(x, scales, centers, out, n4, nTiles);
}